// GeneralLoss_69638599737810
// MI455X (gfx1250) — compile-verified
//
#include <hip/hip_runtime.h>
#include <hip/hip_bf16.h>

// ---------------------------------------------------------------------------
// SSD MultiBox loss for MI455X (gfx1250, wave32).
//   B=16 batches, P=131072 priors, G=64 ground-truth boxes, 2 classes.
// Pipeline (all L2-resident after first touch; HBM ~52MB => ~2.2us @23.3TB/s):
//   K0 init   : zero accumulators + packed per-GT argmax slots
//   K1 match  : stream priors once; GT boxes in LDS; per-prior max/argmax IoU
//               (inter * v_rcp_f32(union) instead of IEEE div ladder),
//               per-GT argmax via packed (iou_bits<<32 | ~prior) u64 atomicMax
//   K2 override: force-match best prior of each GT (bto=2.0, bti=g)
//   K3 loss   : conf, smooth-L1 (positives), CE=lse-gathered; wave-level sums
//               accumulated on the MATRIX CORE via V_WMMA_F32_16X16X4_F32
//               (A = {sl1, ce} per lane, B = column selector)
//   K4 select : exact top-K of loss_c per batch via 4-pass 256-bin radix
//               select on float bit patterns (loss_c >= 0 => monotone)
//   K5 final  : 3 scalars
// ---------------------------------------------------------------------------

#define BATCH   16
#define NPRIOR  131072
#define NGT     64
#define THRESH  0.35f
#define INV_VAR0 10.0f     // 1/0.1
#define INV_VAR1 5.0f      // 1/0.2

typedef __attribute__((ext_vector_type(2))) float v2f;
typedef __attribute__((ext_vector_type(8))) float v8f;

__device__ __forceinline__ float fast_rcp(float x) {
    return __builtin_amdgcn_rcpf(x);   // v_rcp_f32, ~1ulp
}

struct Accum {
    float sl1_sum;            // global smooth-L1 numerator
    float pos_cls_sum;        // global sum of (lse-gathered) over positives
    int   npos_b[BATCH];      // positives per batch
    float topk_sum_b[BATCH];  // hard-negative CE sum per batch
};

// ------------------------------ K0: init -----------------------------------
__global__ void k_init(unsigned long long* __restrict__ bestg, Accum* acc) {
    int t = blockIdx.x * blockDim.x + threadIdx.x;
    if (t < BATCH * NGT) bestg[t] = 0ull;
    if (t == 0) { acc->sl1_sum = 0.f; acc->pos_cls_sum = 0.f; }
    if (t < BATCH) { acc->npos_b[t] = 0; acc->topk_sum_b[t] = 0.f; }
}

// ------------------------------ K1: match ----------------------------------
__global__ __launch_bounds__(256) void k_match(
        const float* __restrict__ prior,    // (P,4) cx,cy,w,h
        const float* __restrict__ targets,  // (B,G,5) y1,x1,y2,x2,label
        float* __restrict__ bto, int* __restrict__ bti,
        unsigned long long* __restrict__ bestg) {
    __shared__ float gx1[NGT], gy1[NGT], gx2[NGT], gy2[NGT], garea[NGT];
    __shared__ unsigned long long lbest[NGT];
    const int b = blockIdx.y;
    const int t = threadIdx.x;
    if (t < NGT) {
        const float* tg = targets + (size_t)(b * NGT + t) * 5;
        float y1 = tg[0], x1 = tg[1], y2 = tg[2], x2 = tg[3];  // un-permute
        gx1[t] = x1; gy1[t] = y1; gx2[t] = x2; gy2[t] = y2;
        garea[t] = (x2 - x1) * (y2 - y1);
        lbest[t] = 0ull;
    }
    __syncthreads();

    const int p = blockIdx.x * 256 + t;
    const float* pr = prior + (size_t)p * 4;
    float cx = pr[0], cy = pr[1], w = pr[2], h = pr[3];
    float px1 = cx - 0.5f * w, py1 = cy - 0.5f * h;
    float px2 = cx + 0.5f * w, py2 = cy + 0.5f * h;
    float parea = (px2 - px1) * (py2 - py1);

    float best = -1.0f; int bg = 0;
    #pragma unroll 4
    for (int g = 0; g < NGT; ++g) {
        float tlx = fmaxf(gx1[g], px1), tly = fmaxf(gy1[g], py1);
        float brx = fminf(gx2[g], px2), bry = fminf(gy2[g], py2);
        float iw = fmaxf(brx - tlx, 0.f), ih = fmaxf(bry - tly, 0.f);
        float inter = iw * ih;
        float iou = inter * fast_rcp(garea[g] + parea - inter);  // v_rcp_f32
        if (iou > best) { best = iou; bg = g; }   // strict > == lowest-index tie
        // per-GT argmax: pack value + complemented index (max => lowest prior)
        unsigned long long pack =
            ((unsigned long long)__float_as_uint(iou) << 32) |
            (unsigned)(0xFFFFFFFFu - (unsigned)p);
        if (pack > lbest[g]) atomicMax(&lbest[g], pack);   // DS_MAX_U64
    }
    bto[(size_t)b * NPRIOR + p] = best;
    bti[(size_t)b * NPRIOR + p] = bg;
    __syncthreads();
    if (t < NGT) atomicMax(&bestg[b * NGT + t], lbest[t]); // GLOBAL_ATOMIC_MAX_U64
}

// ---------------------------- K2: override ---------------------------------
__global__ void k_override(const unsigned long long* __restrict__ bestg,
                           float* __restrict__ bto, int* __restrict__ bti) {
    int b = blockIdx.x;
    if (threadIdx.x != 0) return;
    for (int g = 0; g < NGT; ++g) {           // ascending: last write wins
        unsigned long long pk = bestg[b * NGT + g];
        unsigned p = 0xFFFFFFFFu - (unsigned)(pk & 0xFFFFFFFFull);
        bto[(size_t)b * NPRIOR + p] = 2.0f;
        bti[(size_t)b * NPRIOR + p] = (int)g;
    }
}

// ------------------------------ K3: loss -----------------------------------
#define K3_ITER 4
__global__ __launch_bounds__(256) void k_loss(
        const float* __restrict__ loc_preds, const float* __restrict__ cls_preds,
        const float* __restrict__ prior,     const float* __restrict__ targets,
        const float* __restrict__ bto,       const int* __restrict__ bti,
        float* __restrict__ lossc, Accum* acc) {
    __shared__ float gx1[NGT], gy1[NGT], gx2[NGT], gy2[NGT], glab[NGT];
    __shared__ float red_sl1, red_cls;
    __shared__ int   red_np;
    const int b = blockIdx.y;
    const int t = threadIdx.x;
    if (t < NGT) {
        const float* tg = targets + (size_t)(b * NGT + t) * 5;
        gx1[t] = tg[1]; gy1[t] = tg[0]; gx2[t] = tg[3]; gy2[t] = tg[2];
        glab[t] = tg[4];
    }
    if (t == 0) { red_sl1 = 0.f; red_cls = 0.f; red_np = 0; }
    __syncthreads();

    const int lane = t & 31;
    // Column-selector B for V_WMMA_F32_16X16X4_F32 (B is 4x16, K rows striped:
    // VGPR0 lanes0-15=K0 / lanes16-31=K2, VGPR1 K1/K3). B[k,0]=1 for even k,
    // B[k,1]=1 for odd k  =>  D[:,0]=sum(sl1), D[:,1]=sum(ce).
    v2f bsel;
    bsel.x = (lane == 0 || lane == 16) ? 1.f : 0.f;
    bsel.y = (lane == 1 || lane == 17) ? 1.f : 0.f;
    v8f cacc = {};
    int npos_l = 0;

    for (int it = 0; it < K3_ITER; ++it) {
        int p = (blockIdx.x * K3_ITER + it) * 256 + t;   // full coverage, no tail
        size_t idx = (size_t)b * NPRIOR + p;
        float ov = bto[idx];
        int   g  = bti[idx];
        float conf = (ov < THRESH) ? 0.f : glab[g];
        bool  pos  = conf > 0.f;

        // classification CE = lse - gathered (streamed once: NT loads)
        const float* cp = cls_preds + idx * 2;
        float c0 = __builtin_nontemporal_load(cp);
        float c1 = __builtin_nontemporal_load(cp + 1);
        float mx = fmaxf(c0, c1);
        float lse = mx + __logf(__expf(c0 - mx) + __expf(c1 - mx));  // v_exp/v_log
        int lbl = (int)conf;
        float gathered = lbl ? c1 : c0;
        float lc = lse - gathered;
        lossc[idx] = pos ? 0.f : lc;            // zeroed at positives (ref semantics)
        float cls_contrib = pos ? lc : 0.f;

        // localization smooth-L1 (encode GT against prior); rcp instead of div
        const float* pr = prior + (size_t)p * 4;
        float pcx = pr[0], pcy = pr[1], pw = pr[2], ph = pr[3];
        float invw = fast_rcp(pw), invh = fast_rcp(ph);
        float mx1 = gx1[g], my1 = gy1[g], mx2 = gx2[g], my2 = gy2[g];
        float tx = ((mx1 + mx2) * 0.5f - pcx) * (INV_VAR0 * invw);
        float ty = ((my1 + my2) * 0.5f - pcy) * (INV_VAR0 * invh);
        float tw = __logf((mx2 - mx1) * invw) * INV_VAR1;
        float th = __logf((my2 - my1) * invh) * INV_VAR1;
        const float* lp = loc_preds + idx * 4;
        float d0 = __builtin_nontemporal_load(lp)     - tx;
        float d1 = __builtin_nontemporal_load(lp + 1) - ty;
        float d2 = __builtin_nontemporal_load(lp + 2) - tw;
        float d3 = __builtin_nontemporal_load(lp + 3) - th;
        float s = 0.f;
        {
            float ad = fabsf(d0); s += (ad < 1.f) ? 0.5f * d0 * d0 : ad - 0.5f;
            ad = fabsf(d1);       s += (ad < 1.f) ? 0.5f * d1 * d1 : ad - 0.5f;
            ad = fabsf(d2);       s += (ad < 1.f) ? 0.5f * d2 * d2 : ad - 0.5f;
            ad = fabsf(d3);       s += (ad < 1.f) ? 0.5f * d3 * d3 : ad - 0.5f;
        }
        float sl1_contrib = pos ? s : 0.f;
        npos_l += pos ? 1 : 0;

        // Matrix-core accumulation (uniform control flow => EXEC all ones)
        v2f a; a.x = sl1_contrib; a.y = cls_contrib;
        cacc = __builtin_amdgcn_wmma_f32_16x16x4_f32(
            /*neg_a=*/false, a, /*neg_b=*/false, bsel,
            /*c_mod=*/(short)0, cacc, /*reuse_a=*/false, /*reuse_b=*/false);
    }

    // D[m,n] layout: lane n holds column n (rows 0-7 in c[0..7] for lanes<16,
    // rows 8-15 for lanes>=16). colsum(lane0)+colsum(lane16) = total sl1 etc.
    float colsum = 0.f;
    #pragma unroll
    for (int i = 0; i < 8; ++i) colsum += cacc[i];
    float sl1_wave = __shfl(colsum, 0, 32) + __shfl(colsum, 16, 32);
    float cls_wave = __shfl(colsum, 1, 32) + __shfl(colsum, 17, 32);
    #pragma unroll
    for (int off = 16; off; off >>= 1) npos_l += __shfl_xor(npos_l, off, 32);

    if (lane == 0) {
        atomicAdd(&red_sl1, sl1_wave);
        atomicAdd(&red_cls, cls_wave);
        atomicAdd(&red_np, npos_l);
    }
    __syncthreads();
    if (t == 0) {
        atomicAdd(&acc->sl1_sum, red_sl1);
        atomicAdd(&acc->pos_cls_sum, red_cls);
        atomicAdd(&acc->npos_b[b], red_np);
    }
}

// --------------------- K4: exact top-K radix select ------------------------
__global__ __launch_bounds__(256) void k_select(const float* __restrict__ lossc,
                                                Accum* acc) {
    __shared__ unsigned hist[256];
    __shared__ unsigned s_prefix;
    __shared__ int s_kr;
    __shared__ float s_part[8];
    const int b = blockIdx.x;
    const int t = threadIdx.x;
    const int np = acc->npos_b[b];
    long long K = 3LL * np; if (K > NPRIOR - 1) K = NPRIOR - 1;
    if (K <= 0) { if (t == 0) acc->topk_sum_b[b] = 0.f; return; }

    const float* lc = lossc + (size_t)b * NPRIOR;
    unsigned prefix = 0; int kr = (int)K;

    for (int pass = 0; pass < 4; ++pass) {       // MSB-first byte radix
        const int shift = 24 - pass * 8;
        const unsigned hmask = (pass == 0) ? 0u : (0xFFFFFFFFu << (shift + 8));
        hist[t] = 0;
        __syncthreads();
        for (int p = t; p < NPRIOR; p += 256) {
            __builtin_prefetch(lc + p + 2048, 0, 0);   // global_prefetch_b8
            unsigned v = __float_as_uint(lc[p]);       // loss_c >= 0: monotone bits
            if ((v & hmask) == prefix) atomicAdd(&hist[(v >> shift) & 255], 1u);
        }
        __syncthreads();
        if (t == 0) {
            unsigned cum = 0;
            for (int bin = 255; bin >= 0; --bin) {
                unsigned c = hist[bin];
                if (cum + c >= (unsigned)kr) {
                    s_prefix = prefix | ((unsigned)bin << shift);
                    s_kr = kr - (int)cum;
                    break;
                }
                cum += c;
            }
        }
        __syncthreads();
        prefix = s_prefix; kr = s_kr;
    }

    const float tau = __uint_as_float(prefix);
    float part = 0.f;
    for (int p = t; p < NPRIOR; p += 256) {
        float v = lc[p];
        part += (v > tau) ? v : 0.f;
    }
    #pragma unroll
    for (int off = 16; off; off >>= 1) part += __shfl_xor(part, off, 32);
    if ((t & 31) == 0) s_part[t >> 5] = part;
    __syncthreads();
    if (t == 0) {
        float sum = 0.f;
        for (int w = 0; w < 8; ++w) sum += s_part[w];
        acc->topk_sum_b[b] = sum + (float)kr * tau;   // ties at threshold
    }
}

// ----------------------------- K5: finalize --------------------------------
__global__ void k_final(const Accum* __restrict__ acc, float* __restrict__ out) {
    if (threadIdx.x != 0 || blockIdx.x != 0) return;
    long long npos = 0, maskc = 0; float topk = 0.f;
    for (int b = 0; b < BATCH; ++b) {
        int np = acc->npos_b[b];
        npos += np;
        long long K = 3LL * np; if (K > NPRIOR - 1) K = NPRIOR - 1;
        maskc += np + K;
        topk += acc->topk_sum_b[b];
    }
    long long dl = npos * 4; if (dl < 1) dl = 1;
    float loss_loc = acc->sl1_sum / (float)dl;
    long long dc = maskc; if (dc < 1) dc = 1;
    float loss_cls = (acc->pos_cls_sum + topk) / (float)dc;
    float loss = (loss_cls + 1.0f * loss_loc) / (float)npos;
    out[0] = loss; out[1] = loss_loc; out[2] = loss_cls;
}

// ----------------------------- launcher ------------------------------------
extern "C" void kernel_launch(void* const* d_in, const int* in_sizes, int n_in,
                              void* d_out, int out_size, void* d_ws, size_t ws_size,
                              hipStream_t stream) {
    const float* loc_preds = (const float*)d_in[0];   // (B,P,4)
    const float* cls_preds = (const float*)d_in[1];   // (B,P,2)
    const float* priorbox  = (const float*)d_in[2];   // (P,4)
    const float* targets   = (const float*)d_in[3];   // (B,G,5)
    float* out = (float*)d_out;

    char* ws = (char*)d_ws;
    const size_t bp = (size_t)BATCH * NPRIOR;
    float*              bto   = (float*)ws;                       ws += bp * 4;
    int*                bti   = (int*)ws;                         ws += bp * 4;
    float*              lossc = (float*)ws;                       ws += bp * 4;
    unsigned long long* bestg = (unsigned long long*)ws;          ws += (size_t)BATCH * NGT * 8;
    Accum*              acc   = (Accum*)ws;

    k_init<<<4, 256, 0, stream>>>(bestg, acc);
    k_match<<<dim3(NPRIOR / 256, BATCH), 256, 0, stream>>>(priorbox, targets,
                                                           bto, bti, bestg);
    k_override<<<BATCH, 32, 0, stream>>>(bestg, bto, bti);
    k_loss<<<dim3(NPRIOR / (256 * K3_ITER), BATCH), 256, 0, stream>>>(
        loc_preds, cls_preds, priorbox, targets, bto, bti, lossc, acc);
    k_select<<<BATCH, 256, 0, stream>>>(lossc, acc);
    k_final<<<1, 32, 0, stream>>>(acc, out);
}